// GraphConvolution_66752381714534
// MI455X (gfx1250) — compile-verified
//
#include <hip/hip_runtime.h>

typedef float v2f __attribute__((ext_vector_type(2)));
typedef float v4f __attribute__((ext_vector_type(4)));
typedef float v8f __attribute__((ext_vector_type(8)));

#define IN_DIM   256
#define OUT_DIM  128
#define KCHUNK   64
#define LDS_STRIDE 66   // even (keeps 8B alignment for b64 LDS loads), padded vs 64 to spread banks

// ---------------------------------------------------------------------------
// Dense GEMM: pre[n_nodes][128] = x[n_nodes][256] @ W[256][128]  (fp32 WMMA)
// One wave32 -> 16-row strip x all 8 N-tiles (16x128 output, 8x v8f acc).
// W staged transposed in LDS per 64-wide K chunk, shared by 8 waves/block.
// ---------------------------------------------------------------------------
__global__ __launch_bounds__(256) void gcn_gemm_wmma(
    const float* __restrict__ x, const float* __restrict__ W,
    float* __restrict__ pre, int n_tiles)
{
    __shared__ float ldsW[OUT_DIM * LDS_STRIDE];   // 128*66*4 = 33792 B

    const int tid  = threadIdx.x;
    const int wave = tid >> 5;
    const int lane = tid & 31;
    const int M    = lane & 15;   // row within 16-row A strip / col within N-tile
    const int half = lane >> 4;   // selects K pair {0,1} vs {2,3} per WMMA step
    const int tile = blockIdx.x * 8 + wave;
    const bool active = tile < n_tiles;
    const int row0 = tile * 16;

    v8f acc[8] = {};

    for (int kb = 0; kb < IN_DIM; kb += KCHUNK) {
        __syncthreads();
        // Cooperative transpose-load of W[kb..kb+63][0..127] into LDS:
        // ldsW[c][kk] = W[kb+kk][c]; LDS writes are bank-conflict free.
        for (int idx = tid; idx < OUT_DIM * KCHUNK; idx += 256) {
            int c  = idx >> 6;      // 0..127
            int kk = idx & 63;      // 0..63
            ldsW[c * LDS_STRIDE + kk] = W[(size_t)(kb + kk) * OUT_DIM + c];
        }
        __syncthreads();

        if (active) {
            // A fragment base: lane reads x[row0+M][kb + k2 + 2*half .. +1]
            const float* xrow = x + (size_t)(row0 + M) * IN_DIM + kb + 2 * half;
            #pragma unroll 4
            for (int k2 = 0; k2 < KCHUNK; k2 += 4) {
                v2f a = *(const v2f*)(xrow + k2);               // global_load_b64
                #pragma unroll
                for (int n = 0; n < 8; ++n) {
                    const float* bp =
                        &ldsW[(n * 16 + M) * LDS_STRIDE + k2 + 2 * half];
                    v2f b = *(const v2f*)bp;                     // ds_load_b64
                    acc[n] = __builtin_amdgcn_wmma_f32_16x16x4_f32(
                        false, a, false, b, (short)0, acc[n], false, false);
                }
            }
        }
    }

    if (active) {
        // C/D layout: VGPR v -> M = v + 8*half, N = lane&15 (per ISA table)
        #pragma unroll
        for (int n = 0; n < 8; ++n) {
            #pragma unroll
            for (int v = 0; v < 8; ++v) {
                int r = row0 + v + 8 * half;
                pre[(size_t)r * OUT_DIM + n * 16 + M] = acc[n][v];
            }
        }
    }
}

// ---------------------------------------------------------------------------
// Zero-fill out (needed every call: atomics accumulate into it)
// ---------------------------------------------------------------------------
__global__ __launch_bounds__(256) void gcn_zero(float* __restrict__ out, int n4)
{
    int i = blockIdx.x * 256 + threadIdx.x;
    if (i < n4) {
        v4f z = {};
        *(v4f*)(out + (size_t)i * 4) = z;
    }
}

// ---------------------------------------------------------------------------
// COO scatter: out[row] += val * pre[col]; 32 lanes per edge, 4 cols/lane.
// pre gather is one b128 per lane; adds via global_atomic_add_f32.
// ---------------------------------------------------------------------------
__global__ __launch_bounds__(256) void gcn_scatter(
    const float* __restrict__ pre, const int* __restrict__ row,
    const int* __restrict__ col, const float* __restrict__ vals,
    float* __restrict__ out, int n_edges)
{
    long long gid = (long long)blockIdx.x * 256 + threadIdx.x;
    int e = (int)(gid >> 5);
    if (e >= n_edges) return;
    int c4 = ((int)gid & 31) * 4;

    int   r = row[e];
    int   c = col[e];
    float v = vals[e];

    v4f p = *(const v4f*)(pre + (size_t)c * OUT_DIM + c4);
    float* op = out + (size_t)r * OUT_DIM + c4;
    unsafeAtomicAdd(op + 0, v * p.x);
    unsafeAtomicAdd(op + 1, v * p.y);
    unsafeAtomicAdd(op + 2, v * p.z);
    unsafeAtomicAdd(op + 3, v * p.w);
}

// ---------------------------------------------------------------------------
// In-place ReLU
// ---------------------------------------------------------------------------
__global__ __launch_bounds__(256) void gcn_relu(float* __restrict__ out, int n4)
{
    int i = blockIdx.x * 256 + threadIdx.x;
    if (i < n4) {
        v4f t = *(v4f*)(out + (size_t)i * 4);
        t.x = fmaxf(t.x, 0.f);
        t.y = fmaxf(t.y, 0.f);
        t.z = fmaxf(t.z, 0.f);
        t.w = fmaxf(t.w, 0.f);
        *(v4f*)(out + (size_t)i * 4) = t;
    }
}

extern "C" void kernel_launch(void* const* d_in, const int* in_sizes, int n_in,
                              void* d_out, int out_size, void* d_ws, size_t ws_size,
                              hipStream_t stream)
{
    const float* x    = (const float*)d_in[0];   // [N, 256]
    const float* W    = (const float*)d_in[1];   // [256, 128]
    const int*   row  = (const int*)d_in[2];     // [E]
    const int*   col  = (const int*)d_in[3];     // [E]
    const float* vals = (const float*)d_in[4];   // [E]
    float* out = (float*)d_out;                  // [N, 128]
    float* pre = (float*)d_ws;                   // scratch: N*128 floats

    const int n_nodes = in_sizes[0] / IN_DIM;
    const int n_edges = in_sizes[2];
    const int n_tiles = (n_nodes + 15) / 16;
    const int out_n4  = (n_nodes * OUT_DIM) / 4;

    gcn_zero<<<(out_n4 + 255) / 256, 256, 0, stream>>>(out, out_n4);

    gcn_gemm_wmma<<<(n_tiles + 7) / 8, 256, 0, stream>>>(x, W, pre, n_tiles);

    long long sc_threads = (long long)n_edges * 32;
    int sc_blocks = (int)((sc_threads + 255) / 256);
    gcn_scatter<<<sc_blocks, 256, 0, stream>>>(pre, row, col, vals, out, n_edges);

    gcn_relu<<<(out_n4 + 255) / 256, 256, 0, stream>>>(out, out_n4);
}